// GaussianRenderer2D_90331752169585
// MI455X (gfx1250) — compile-verified
//
#include <hip/hip_runtime.h>
#include <hip/hip_bf16.h>

typedef __attribute__((ext_vector_type(2))) float v2f;
typedef __attribute__((ext_vector_type(8))) float v8f;

#define IMG_S   128
#define NGAUSS  2048
#define NBATCH  8
#define NCHAN   32
#define CHUNK_G 128
#define GEPS    1e-6f

// sqrt(0.5*log2(e)) : folding exp(-0.5*t) = exp2(-(sqrt(k)*x)^2 - (sqrt(k)*y)^2)
#define SQRT_HALF_LOG2E 0.84932180553116060f

__global__ __launch_bounds__(256)
void gs2d_render_kernel(const float* __restrict__ position,
                        const float* __restrict__ scale,
                        const float* __restrict__ rotation,
                        const float* __restrict__ opacity,
                        const float* __restrict__ feature,
                        float* __restrict__ out)
{
    // Per-gaussian folded params: {px,py,A,B} {C,D,G,pad} -> 2 float4 each
    //   u = A*dx + B*dy ; v = C*dx + D*dy ; w = exp2(G - u*u - v*v)
    __shared__ float4 s_par[CHUNK_G * 2];                 // 4 KB
    __shared__ float4 s_feat[CHUNK_G * NCHAN / 4];        // 16 KB

    const int tid  = threadIdx.x;
    const int lane = tid & 31;
    const int wv   = tid >> 5;           // 8 waves -> 8 x 16-pixel tiles = 1 row
    const int b    = blockIdx.x >> 7;    // blockIdx.x = b*128 + h
    const int h    = blockIdx.x & 127;

    const int   col = (wv << 4) + (lane & 15);            // pixel x for this lane
    const float gx  = (float)col * (1.0f / 127.0f);       // linspace(0,1,128)
    const float gy  = (float)h   * (1.0f / 127.0f);

    v8f d0 = {};          // channels [0..15]  x 16 pixels
    v8f d1 = {};          // channels [16..31] x 16 pixels
    float o_acc = 0.0f;   // per-lane partial of omap (2 K-halves folded later)

    const float* s_featf = (const float*)s_feat;

    for (int n0 = 0; n0 < NGAUSS; n0 += CHUNK_G) {
        __syncthreads();
        // ---- stage per-gaussian folded params (cold path) ----
        if (tid < CHUNK_G) {
            const int gi = b * NGAUSS + n0 + tid;
            float px  = position[gi * 2 + 0];
            float py  = position[gi * 2 + 1];
            float sx  = fmaxf(scale[gi * 2 + 0], GEPS);
            float sy  = fmaxf(scale[gi * 2 + 1], GEPS);
            float ang = rotation[gi];
            float ca  = cosf(ang);
            float sa  = sinf(ang);
            float op  = opacity[gi];
            float rxs = SQRT_HALF_LOG2E / sx;
            float rys = SQRT_HALF_LOG2E / sy;
            float A = ca * rxs, B = sa * rxs;
            float C = -sa * rys, D = ca * rys;
            float G = __builtin_amdgcn_logf(op);   // v_log_f32 == log2
            s_par[tid * 2 + 0] = make_float4(px, py, A, B);
            s_par[tid * 2 + 1] = make_float4(C, D, G, 0.0f);
        }
        // ---- stage feature chunk (128 gaussians x 32 ch) ----
        {
            const float4* src = (const float4*)(feature + (size_t)(b * NGAUSS + n0) * NCHAN);
            #pragma unroll
            for (int i = 0; i < (CHUNK_G * NCHAN / 4) / 256; ++i)
                s_feat[tid + i * 256] = src[tid + i * 256];
        }
        __syncthreads();

        // ---- prefetch next feature chunk into cache (global_prefetch_b8) ----
        if (n0 + CHUNK_G < NGAUSS) {
            const char* nsrc = (const char*)(feature + (size_t)(b * NGAUSS + n0 + CHUNK_G) * NCHAN);
            __builtin_prefetch(nsrc + tid * 64, 0, 1);   // 256 threads x 64B = 16 KB
        }

        // ---- K loop: 4 gaussians per WMMA step ----
        #pragma unroll 4
        for (int k0 = 0; k0 < CHUNK_G; k0 += 4) {
            const int kb = k0 + ((lane >> 4) << 1);   // lanes 16..31 take K+2,K+3

            // weight for gaussian kb
            float4 p0 = s_par[kb * 2 + 0];
            float4 q0 = s_par[kb * 2 + 1];
            float dx0 = gx - p0.x, dy0 = gy - p0.y;
            float u0 = fmaf(p0.w, dy0, p0.z * dx0);
            float v0 = fmaf(q0.y, dy0, q0.x * dx0);
            float a0e = fmaf(-u0, u0, fmaf(-v0, v0, q0.z));
            float w0 = __builtin_amdgcn_exp2f(a0e);   // bare v_exp_f32

            // weight for gaussian kb+1
            float4 p1 = s_par[(kb + 1) * 2 + 0];
            float4 q1 = s_par[(kb + 1) * 2 + 1];
            float dx1 = gx - p1.x, dy1 = gy - p1.y;
            float u1 = fmaf(p1.w, dy1, p1.z * dx1);
            float v1 = fmaf(q1.y, dy1, q1.x * dx1);
            float a1e = fmaf(-u1, u1, fmaf(-v1, v1, q1.z));
            float w1 = __builtin_amdgcn_exp2f(a1e);

            o_acc += w0 + w1;

            v2f bm; bm.x = w0; bm.y = w1;   // B: 4x16 weights (K x pixels)

            const int m = lane & 15;        // channel within C-tile
            v2f a0, a1;                      // A: 16x4 features (C x K)
            a0.x = s_featf[kb * NCHAN + m];
            a0.y = s_featf[(kb + 1) * NCHAN + m];
            a1.x = s_featf[kb * NCHAN + 16 + m];
            a1.y = s_featf[(kb + 1) * NCHAN + 16 + m];

            d0 = __builtin_amdgcn_wmma_f32_16x16x4_f32(false, a0, false, bm,
                                                       (short)0, d0, false, false);
            d1 = __builtin_amdgcn_wmma_f32_16x16x4_f32(false, a1, false, bm,
                                                       (short)0, d1, false, false);
        }
    }

    // ---- epilogue: fold omap halves, normalize, write ----
    float o_tot = o_acc + __shfl_xor(o_acc, 16, 32);
    float invd  = 1.0f / fmaxf(o_tot, GEPS);

    const int    mo      = (lane >> 4) * 8;                 // D row offset per lane-half
    const size_t pixbase = (size_t)h * IMG_S + col;
    const size_t bbase   = (size_t)b * (NCHAN + 1) * IMG_S * IMG_S;

    #pragma unroll
    for (int r = 0; r < 8; ++r) {
        const int c0 = mo + r;
        out[bbase + (size_t)c0        * (IMG_S * IMG_S) + pixbase] = d0[r] * invd;
        out[bbase + (size_t)(c0 + 16) * (IMG_S * IMG_S) + pixbase] = d1[r] * invd;
    }
    if (lane < 16) {
        out[bbase + (size_t)NCHAN * (IMG_S * IMG_S) + pixbase] =
            fminf(fmaxf(o_tot, 0.0f), 1.0f);
    }
}

extern "C" void kernel_launch(void* const* d_in, const int* in_sizes, int n_in,
                              void* d_out, int out_size, void* d_ws, size_t ws_size,
                              hipStream_t stream) {
    const float* position = (const float*)d_in[0];
    const float* scale    = (const float*)d_in[1];
    const float* rotation = (const float*)d_in[2];
    const float* opacity  = (const float*)d_in[3];
    const float* feature  = (const float*)d_in[4];
    float* out = (float*)d_out;

    dim3 grid(NBATCH * IMG_S);   // 1024 blocks: (batch, row)
    dim3 block(256);             // 8 waves x 16-pixel tiles = one row
    gs2d_render_kernel<<<grid, block, 0, stream>>>(position, scale, rotation,
                                                   opacity, feature, out);
}